// VectorQuantGroup_4406636446031
// MI455X (gfx1250) — compile-verified
//
#include <hip/hip_runtime.h>
#include <math.h>

typedef __attribute__((ext_vector_type(2))) float v2f;
typedef __attribute__((ext_vector_type(8))) float v8f;

#define T_TOTAL        65536
#define KCLS           512
#define VDIM           128
#define PSZ            16
#define NGROUPS        32
#define ROWS_PER_BLOCK 128
#define ROWS_PER_WAVE  16
#define NWAVES         8
#define EMB_STRIDE     130   // class-major row stride (floats); 130 ≡ 2 mod 64 banks

// d_out layout (floats), concatenated tuple in return order
#define OFF_VQ   (T_TOTAL * VDIM)
#define OFF_ENC  (OFF_VQ + T_TOTAL)
#define OFF_ENT  (OFF_ENC + T_TOTAL)
#define OFF_IA   (OFF_ENT + 1)
#define OFF_IG   (OFF_IA + T_TOTAL)

__global__ void vq_zero_hist(float* hist) { hist[threadIdx.x] = 0.0f; }

__global__ __launch_bounds__(256, 1)
void vq_main(const float* __restrict__ x, const float* __restrict__ emb,
             float* __restrict__ out, float* __restrict__ hist)
{
  __shared__ float embC[KCLS * EMB_STRIDE];          // embC[class][v], padded rows
  __shared__ float e2s[KCLS];
  __shared__ float x2s[ROWS_PER_BLOCK];
  __shared__ float bestGSum[ROWS_PER_BLOCK];
  __shared__ int   bestG[ROWS_PER_BLOCK];
  __shared__ float bestAD[ROWS_PER_BLOCK];
  __shared__ int   bestAI[ROWS_PER_BLOCK];
  __shared__ float bestDist[ROWS_PER_BLOCK * PSZ];
  __shared__ float tileD[NWAVES * 16 * 17];

  const int tid  = threadIdx.x;
  const int lane = tid & 31;
  const int wv   = tid >> 5;
  const int blockRow0 = blockIdx.x * ROWS_PER_BLOCK;

  // ---- Phase 0: stage embedding (class-major, padded) in LDS ----
  for (int i = tid; i < KCLS * (VDIM / 4); i += 256) {
    const int c  = i >> 5;           // class
    const int v0 = (i & 31) << 2;    // feature base
    const float4 e = ((const float4*)emb)[i];
    float* row = &embC[c * EMB_STRIDE + v0];
    row[0] = e.x; row[1] = e.y; row[2] = e.z; row[3] = e.w;
  }
  __syncthreads();

  // e2 per class
  for (int k = tid; k < KCLS; k += 256) {
    float s = 0.0f;
    const float* row = &embC[k * EMB_STRIDE];
    for (int v = 0; v < VDIM; ++v) { const float ev = row[v]; s += ev * ev; }
    e2s[k] = s;
  }

  // x2 per row (wave owns 16 rows)
  for (int r = 0; r < ROWS_PER_WAVE; ++r) {
    const int rl = wv * ROWS_PER_WAVE + r;
    const float* xr = x + (size_t)(blockRow0 + rl) * VDIM;
    float s = 0.0f;
    #pragma unroll
    for (int vv = 0; vv < 4; ++vv) { const float xv = xr[lane + 32 * vv]; s += xv * xv; }
    s += __shfl_xor(s, 16);
    s += __shfl_xor(s, 8);
    s += __shfl_xor(s, 4);
    s += __shfl_xor(s, 2);
    s += __shfl_xor(s, 1);
    if (lane == 0) x2s[rl] = s;
  }

  if (tid < ROWS_PER_BLOCK) {
    bestGSum[tid] = 3.0e38f; bestG[tid] = 0;
    bestAD[tid]   = 3.0e38f; bestAI[tid] = 0;
  }
  __syncthreads();

  // ---- Phase 1: load A fragments (V_WMMA_F32_16X16X4_F32 layout) ----
  // lanes 0-15: row = lane, K pair {4s, 4s+1}; lanes 16-31: row = lane-16, {4s+2, 4s+3}
  const int half = lane >> 4;
  const int l15  = lane & 15;
  const float* xrow = x + (size_t)(blockRow0 + wv * ROWS_PER_WAVE + l15) * VDIM;
  __builtin_prefetch(xrow, 0, 3);          // global_prefetch_b8: keep row warm for phase 2
  v2f afrag[32];
  #pragma unroll
  for (int s = 0; s < 32; ++s)
    afrag[s] = *(const v2f*)(xrow + 4 * s + 2 * half);

  float* tw = &tileD[wv * 16 * 17];

  // ---- 32 column tiles; each 16-wide tile == one group ----
  for (int g = 0; g < NGROUPS; ++g) {
    const int n0 = g * 16;
    const float e2c = e2s[n0 + l15];
    // B fragment base: column (class) n0+l15, feature offset 2*half.
    // All 32 K-steps are immediate DS offsets (16*s bytes) off this one base.
    const float* bptr = &embC[(n0 + l15) * EMB_STRIDE + 2 * half];
    v8f acc = {0.f, 0.f, 0.f, 0.f, 0.f, 0.f, 0.f, 0.f};
    #pragma unroll
    for (int s = 0; s < 32; ++s) {
      const v2f b = *(const v2f*)(bptr + 4 * s);   // ds_load_b64, offset 16*s
      acc = __builtin_amdgcn_wmma_f32_16x16x4_f32(false, afrag[s], false, b,
                                                  (short)0, acc, false, false);
    }
    // d = sqrt(max(x2 - 2*xe + e2, 0)); C layout: lane<16 -> M=r, lane>=16 -> M=r+8, N=lane&15
    #pragma unroll
    for (int r = 0; r < 8; ++r) {
      const int m  = r + 8 * half;
      const float xx = x2s[wv * ROWS_PER_WAVE + m];
      const float d2 = xx - 2.0f * acc[r] + e2c;
      tw[m * 17 + l15] = sqrtf(fmaxf(d2, 0.0f));
    }
    // tileD is per-wave; LDS ops of one wave are in-order -> only a scheduling fence needed
    __builtin_amdgcn_wave_barrier();
    if (lane < 16) {
      const int rl = wv * ROWS_PER_WAVE + lane;
      float sum = 0.0f, mn = 3.0e38f; int am = 0;
      #pragma unroll
      for (int j = 0; j < 16; ++j) {
        const float dv = tw[lane * 17 + j];
        sum += dv;
        if (dv < mn) { mn = dv; am = j; }
      }
      const float gmean = sum * (1.0f / 16.0f);
      if (gmean < bestGSum[rl]) {
        bestGSum[rl] = gmean; bestG[rl] = g;
        #pragma unroll
        for (int j = 0; j < 16; ++j) bestDist[rl * PSZ + j] = tw[lane * 17 + j];
      }
      if (mn < bestAD[rl]) { bestAD[rl] = mn; bestAI[rl] = n0 + am; }
    }
    __builtin_amdgcn_wave_barrier();
  }

  // ---- Phase 2: outputs (top-16 == whole winning group; weighted sum is order-invariant) ----
  // Gather embC[cls*130 + v]: lanes on consecutive v -> bank = 2*cls + v, conflict-free.
  for (int r = 0; r < ROWS_PER_WAVE; ++r) {
    const int rl = wv * ROWS_PER_WAVE + r;
    const size_t rowG = (size_t)blockRow0 + rl;
    const int bg   = bestG[rl];
    const int cls0 = bg * PSZ;
    float S = 0.0f, mn = 3.0e38f; int jm = 0;
    float a0 = 0.f, a1 = 0.f, a2 = 0.f, a3 = 0.f;
    #pragma unroll
    for (int j = 0; j < PSZ; ++j) {
      const float dv = bestDist[rl * PSZ + j];
      const float wj = 1.0f / dv;
      S += wj;
      if (dv < mn) { mn = dv; jm = j; }
      const float* erow = &embC[(cls0 + j) * EMB_STRIDE];
      a0 += wj * erow[lane     ];
      a1 += wj * erow[lane + 32];
      a2 += wj * erow[lane + 64];
      a3 += wj * erow[lane + 96];
    }
    const float invS = 1.0f / fmaxf(S, 1e-12f);
    const float o0 = a0 * invS, o1 = a1 * invS, o2 = a2 * invS, o3 = a3 * invS;
    const float* xr = x + rowG * VDIM;
    float pen = 0.0f, d;
    d = xr[lane     ] - o0; pen += d * d;
    d = xr[lane + 32] - o1; pen += d * d;
    d = xr[lane + 64] - o2; pen += d * d;
    d = xr[lane + 96] - o3; pen += d * d;
    float* orow = out + rowG * VDIM;
    orow[lane     ] = o0;
    orow[lane + 32] = o1;
    orow[lane + 64] = o2;
    orow[lane + 96] = o3;
    pen += __shfl_xor(pen, 16);
    pen += __shfl_xor(pen, 8);
    pen += __shfl_xor(pen, 4);
    pen += __shfl_xor(pen, 2);
    pen += __shfl_xor(pen, 1);
    if (lane == 0) {
      out[OFF_VQ  + rowG] = pen;
      out[OFF_ENC + rowG] = pen;               // + sum((x-x0)^2) == 0
      out[OFF_IA  + rowG] = (float)bestAI[rl];
      out[OFF_IG  + rowG] = (float)bg;
      atomicAdd(&hist[cls0 + jm], 1.0f);
    }
  }
}

__global__ void vq_entropy(const float* __restrict__ hist, float* __restrict__ out_ent)
{
  __shared__ float red[KCLS];
  const int t = threadIdx.x;
  const float h = hist[t];
  float c = 0.0f;
  if (h > 0.0f) {
    const float pr = h * (1.0f / (float)T_TOTAL);
    c = -pr * logf(pr);
  }
  red[t] = c;
  __syncthreads();
  for (int s = KCLS / 2; s > 0; s >>= 1) {
    if (t < s) red[t] += red[t + s];
    __syncthreads();
  }
  if (t == 0) *out_ent = red[0];
}

extern "C" void kernel_launch(void* const* d_in, const int* in_sizes, int n_in,
                              void* d_out, int out_size, void* d_ws, size_t ws_size,
                              hipStream_t stream)
{
  const float* x   = (const float*)d_in[0];   // (16,4096,1,128) fp32
  const float* emb = (const float*)d_in[1];   // (1,512,128) fp32
  float* out  = (float*)d_out;
  float* hist = (float*)d_ws;                 // 512 floats of scratch

  vq_zero_hist<<<1, KCLS, 0, stream>>>(hist);
  vq_main<<<T_TOTAL / ROWS_PER_BLOCK, 256, 0, stream>>>(x, emb, out, hist);
  vq_entropy<<<1, KCLS, 0, stream>>>(hist, out + OFF_ENT);
}